// BoundaryLoss_730144440967
// MI455X (gfx1250) — compile-verified
//
#include <hip/hip_runtime.h>
#include <hip/hip_bf16.h>
#include <stdint.h>

#define BIG 1e9f

// ---------------- CDNA5 async global->LDS path (guarded) ----------------
#if defined(__has_builtin)
#  if __has_builtin(__builtin_amdgcn_global_load_async_to_lds_b32)
#    define HAVE_ASYNC_LDS 1
#  endif
#endif
#ifndef HAVE_ASYNC_LDS
#  define HAVE_ASYNC_LDS 0
#endif

#if HAVE_ASYNC_LDS
typedef __attribute__((address_space(1))) int GInt;   // global (AS1)
typedef __attribute__((address_space(3))) int LInt;   // LDS (AS3)
__device__ __forceinline__ void wait_async0() {
#if defined(__has_builtin) && __has_builtin(__builtin_amdgcn_s_wait_asynccnt)
  __builtin_amdgcn_s_wait_asynccnt(0);
#else
  asm volatile("s_wait_asynccnt 0" ::: "memory");
#endif
}
#endif

// ---------------- WMMA-based exact f32 block reduction ------------------
// Sum-by-ones with V_WMMA_F32_16X16X4_F32 (f32 A/B, f32 accum => exact f32
// adds, deterministic hardware order). A_vgpr0 = lane value, A_vgpr1 = 0:
//   A[m][0] = val[m], A[m][2] = val[m+16]  => row sum R[m] = val[m]+val[m+16]
// D[m][n] = R[m] for all n. Lanes 0-15 hold rows 0-7, lanes 16-31 rows 8-15;
// summing the 8 D regs + one shfl_xor(16) yields the full 32-lane sum.
typedef __attribute__((ext_vector_type(2))) float v2f;
typedef __attribute__((ext_vector_type(8))) float v8f;

__device__ __forceinline__ float wave_sum_wmma(float val) {
  v2f a; a[0] = val;  a[1] = 0.0f;
  v2f b; b[0] = 1.0f; b[1] = 1.0f;   // whole 4x16 B matrix = 1.0
  v8f c = {};
  v8f d = __builtin_amdgcn_wmma_f32_16x16x4_f32(
      /*neg_a=*/false, a, /*neg_b=*/false, b,
      /*c_mod=*/(short)0, c, /*reuse_a=*/false, /*reuse_b=*/false);
  float s = ((d[0] + d[1]) + (d[2] + d[3])) + ((d[4] + d[5]) + (d[6] + d[7]));
  s += __shfl_xor(s, 16, 32);        // fold row halves -> full wave sum
  return s;
}

// Block-level sum over 256 threads; valid on thread 0 only. Deterministic.
__device__ __forceinline__ float block_sum_256(float val) {
  __shared__ float wsum[8];
  const int tid = threadIdx.x;
  float s = wave_sum_wmma(val);      // all threads participate (EXEC all 1s)
  if ((tid & 31) == 0) wsum[tid >> 5] = s;
  __syncthreads();
  float tot = 0.0f;
  if (tid == 0) {
#pragma unroll
    for (int w = 0; w < 8; ++w) tot += wsum[w];
  }
  return tot;
}

// ---------------- Pass 1: one-hot + EDT along W (contiguous lines) ------
// lines indexed [B][C][D][H]; each block handles 4 lines of 64.
__global__ void edt_pass_w(const int* __restrict__ tgt, float* __restrict__ out) {
  __shared__ float sm[4][64];
  const int tid = threadIdx.x;
  const int l = tid >> 6;        // 0..3 line within block
  const int i = tid & 63;        // output position along W
  const int lineIdx = blockIdx.x * 4 + l;   // [0, 32768)
  const int h = lineIdx & 63;
  const int d = (lineIdx >> 6) & 63;
  const int c = (lineIdx >> 12) & 3;
  const int b = lineIdx >> 14;
  const int* tl = tgt + (((b * 64 + d) * 64 + h) * 64);
  sm[l][i] = (tl[i] == c) ? 0.0f : BIG;
  __syncthreads();
  float acc = BIG;
#pragma unroll 8
  for (int j = 0; j < 64; ++j) {
    float diff = (float)(i - j);
    acc = fminf(acc, fmaf(diff, diff, sm[l][j]));   // == sm + (i-j)^2 exactly
  }
  out[(size_t)lineIdx * 64 + i] = acc;
}

// ---------------- Generic strided EDT pass (in place) -------------------
// View buf as [outer][n=64][inner]; transform along n. Each block owns one
// (outer, 64-wide inner chunk) tile: read set == write set -> in-place safe.
__global__ void edt_pass_strided(float* __restrict__ buf, int stride_n,
                                 int chunks_per_outer, int outer_stride) {
  __shared__ float sm[64][65];   // padded, conflict-free
  const int tid  = threadIdx.x;
  const int t    = tid & 63;     // position within inner chunk (coalesced)
  const int isub = tid >> 6;     // 0..3
  const int chunk = blockIdx.x % chunks_per_outer;
  const int outer = blockIdx.x / chunks_per_outer;
  const size_t base = (size_t)outer * outer_stride + (size_t)chunk * 64;

#if HAVE_ASYNC_LDS
#pragma unroll
  for (int r = 0; r < 16; ++r) {
    const int i = isub + (r << 2);
    __builtin_amdgcn_global_load_async_to_lds_b32(
        (GInt*)(buf + base + (size_t)i * stride_n + t),
        (LInt*)&sm[i][t], 0, 0);
  }
  wait_async0();
#else
#pragma unroll
  for (int r = 0; r < 16; ++r) {
    const int i = isub + (r << 2);
    sm[i][t] = buf[base + (size_t)i * stride_n + t];
  }
#endif
  __syncthreads();

  float acc[16];
#pragma unroll
  for (int k = 0; k < 16; ++k) acc[k] = BIG;
#pragma unroll 4
  for (int j = 0; j < 64; ++j) {
    const float f = sm[j][t];
#pragma unroll
    for (int k = 0; k < 16; ++k) {
      float diff = (float)(isub + (k << 2) - j);
      acc[k] = fminf(acc[k], fmaf(diff, diff, f));
    }
  }
#pragma unroll
  for (int k = 0; k < 16; ++k) {
    const int i = isub + (k << 2);
    buf[base + (size_t)i * stride_n + t] = acc[k];
  }
}

// ---------------- Finalize: softmax * sqrt(dist2), block partial sums ---
__global__ void boundary_loss_partials(const float* __restrict__ pred,
                                       const float* __restrict__ fsq,
                                       float* __restrict__ partials) {
  const int v = blockIdx.x * 256 + threadIdx.x;   // voxel in [B][D][H][W]
  const int b = v >> 18;                          // / 262144
  const int rem = v & 262143;
  const float* pb = pred + ((size_t)b * 4) * 262144 + rem;
  const float* fb = fsq  + ((size_t)b * 4) * 262144 + rem;
  __builtin_prefetch(fb, 0, 0);                   // global_prefetch_b8
  const float p0 = pb[0], p1 = pb[262144], p2 = pb[2 * 262144], p3 = pb[3 * 262144];
  const float m  = fmaxf(fmaxf(p0, p1), fmaxf(p2, p3));
  const float e0 = expf(p0 - m), e1 = expf(p1 - m), e2 = expf(p2 - m), e3 = expf(p3 - m);
  const float inv = 1.0f / (e0 + e1 + e2 + e3);
  float s = e0 * inv * sqrtf(fb[0])
          + e1 * inv * sqrtf(fb[262144])
          + e2 * inv * sqrtf(fb[2 * 262144])
          + e3 * inv * sqrtf(fb[3 * 262144]);

  float tot = block_sum_256(s);                   // v_wmma_f32_16x16x4_f32
  if (threadIdx.x == 0) partials[blockIdx.x] = tot;
}

// ---------------- Deterministic final reduction (1 block) ---------------
__global__ void boundary_loss_reduce(const float* __restrict__ partials,
                                     float* __restrict__ out) {
  float s = 0.0f;
#pragma unroll
  for (int r = 0; r < 8; ++r) s += partials[threadIdx.x + 256 * r];
  float tot = block_sum_256(s);                   // v_wmma_f32_16x16x4_f32
  if (threadIdx.x == 0) out[0] = tot * (1.0f / 2097152.0f);  // mean over B*C*D*H*W
}

extern "C" void kernel_launch(void* const* d_in, const int* in_sizes, int n_in,
                              void* d_out, int out_size, void* d_ws, size_t ws_size,
                              hipStream_t stream) {
  const float* pred = (const float*)d_in[0];   // [2,4,64,64,64] f32
  const int*   tgt  = (const int*)d_in[1];     // [2,64,64,64]   int
  float* out = (float*)d_out;                  // scalar f32
  float* fsq = (float*)d_ws;                   // 2,097,152 f32 (8 MB)
  float* partials = fsq + (size_t)2097152;     // 2048 f32

  // Pass 1: one-hot + EDT along W. 32768 lines, 4/block.
  edt_pass_w<<<8192, 256, 0, stream>>>(tgt, fsq);
  // Pass 2: EDT along H (stride 64). outer = B*C*D = 512, 1 chunk each.
  edt_pass_strided<<<512, 256, 0, stream>>>(fsq, 64, 1, 4096);
  // Pass 3: EDT along D (stride 4096). outer = B*C = 8, 64 chunks each.
  edt_pass_strided<<<512, 256, 0, stream>>>(fsq, 4096, 64, 262144);
  // Finalize: softmax * sqrt(dist2), per-block WMMA partials, deterministic reduce.
  boundary_loss_partials<<<2048, 256, 0, stream>>>(pred, fsq, partials);
  boundary_loss_reduce<<<1, 256, 0, stream>>>(partials, out);
}